// GASConv_76716705841589
// MI455X (gfx1250) — compile-verified
//
#include <hip/hip_runtime.h>

// ---------------------------------------------------------------- constants
#define NU_NODES 50000
#define NV_NODES 50000
#define E_CNT    500000
// feature dims
#define E_IN   64
#define U_IN   128
#define E_OUT  128
#define ATT_D  192   // V_IN + E_IN == U_IN + E_IN

typedef __attribute__((ext_vector_type(2))) float v2f;
typedef __attribute__((ext_vector_type(8))) float v8f;

// ---------------------------------------------------------------- ordered-key float max
// Monotone bijection float -> uint32 so that unsigned compare == float compare.
// Lets us use a single unconditional global_atomic_max_u32 (no sign branch).
__device__ __forceinline__ unsigned f32_key(float x) {
    int b = __float_as_int(x);
    return (unsigned)(b ^ ((b >> 31) | 0x80000000));
}
__device__ __forceinline__ float key_f32(unsigned u) {
    unsigned m = ((int)u < 0) ? 0x80000000u : 0xFFFFFFFFu;
    return __int_as_float((int)(u ^ m));
}
#define NEG_INF_KEY 0x007FFFFFu   // f32_key(-inf)

__global__ __launch_bounds__(256) void fill_u32(unsigned* __restrict__ p, unsigned v, int n) {
    int i = blockIdx.x * blockDim.x + threadIdx.x;
    if (i < n) p[i] = v;
}

// ---------------------------------------------------------------- generic fp32 WMMA GEMM
// out[m, colOff + n] = sum_k Aeff[m,k] * W[k,n] + bias[n]
// Aeff = HAS_S ? A[m,k] * (S[m,k] > 0 ? rcp(S[m,k]) : 0) : A[m,k]
// K, N compile-time -> all load offsets are instruction immediates.
// One wave computes a 16x16 tile via V_WMMA_F32_16X16X4_F32, K stepped by 4.
template <int K, int N, bool HAS_S>
__global__ __launch_bounds__(256) void gemm_wmma_f32(
        const float* __restrict__ A, const float* __restrict__ S,
        const float* __restrict__ W, const float* __restrict__ bias,
        float* __restrict__ out, int M, int ldo, int colOff) {
    const int wave = threadIdx.x >> 5;
    const int lane = threadIdx.x & 31;
    const int tilesN = N >> 4;
    const int tile = blockIdx.x * (blockDim.x >> 5) + wave;
    if (tile >= (M >> 4) * tilesN) return;
    const int mt = tile / tilesN, nt = tile % tilesN;
    const int m0 = mt << 4, n0 = nt << 4;
    const int row = lane & 15;       // M (for A) / N (for B,C) index within tile
    const int kh  = lane >> 4;       // K-half select

    const float* Arow = A + (size_t)(m0 + row) * K + 2 * kh;         // + k (imm)
    const float* Srow = HAS_S ? S + (size_t)(m0 + row) * K + 2 * kh : nullptr;
    const float* Wcol = W + n0 + row + (size_t)(2 * kh) * N;         // + k*N (imm)

    v8f acc = {};
#pragma unroll 8
    for (int k = 0; k < K; k += 4) {
        v2f a, b;
        a[0] = Arow[k];
        a[1] = Arow[k + 1];
        if (HAS_S) {
            // branchless fast normalization by softmax denominator
            const float s0 = Srow[k], s1 = Srow[k + 1];
            const float r0 = s0 > 0.f ? __builtin_amdgcn_rcpf(s0) : 0.f;
            const float r1 = s1 > 0.f ? __builtin_amdgcn_rcpf(s1) : 0.f;
            a[0] *= r0; a[1] *= r1;
        }
        b[0] = Wcol[(size_t)k * N];
        b[1] = Wcol[(size_t)(k + 1) * N];
        acc = __builtin_amdgcn_wmma_f32_16x16x4_f32(false, a, false, b,
                                                    (short)0, acc, false, false);
    }
#pragma unroll
    for (int i = 0; i < 8; ++i) {
        const int m = m0 + i + 8 * kh;
        const int n = n0 + row;
        out[(size_t)m * ldo + colOff + n] = acc[i] + bias[n];
    }
}

// ---------------------------------------------------------------- fused edge linear
// he = e_feat @ e_lin_w + e_lin_b + he_u[u_idx] + he_v[v_idx]   (written once, 256 MB)
__global__ __launch_bounds__(256) void edge_he_wmma(
        const float* __restrict__ ef, const float* __restrict__ Wl,
        const float* __restrict__ bl,
        const float* __restrict__ he_u, const float* __restrict__ he_v,
        const int* __restrict__ u_idx, const int* __restrict__ v_idx,
        float* __restrict__ out) {
    const int wave = threadIdx.x >> 5;
    const int lane = threadIdx.x & 31;
    const int tile = blockIdx.x * (blockDim.x >> 5) + wave;
    const int mt = tile >> 3, nt = tile & 7;      // N = 128 -> 8 n-tiles
    if (mt >= E_CNT / 16) return;
    const int m0 = mt << 4, n0 = nt << 4;
    const int row = lane & 15, kh = lane >> 4;

    const float* Arow = ef + (size_t)(m0 + row) * E_IN + 2 * kh;
    const float* Wcol = Wl + n0 + row + (size_t)(2 * kh) * E_OUT;

    v8f acc = {};
#pragma unroll
    for (int k = 0; k < E_IN; k += 4) {
        v2f a, b;
        a[0] = Arow[k];
        a[1] = Arow[k + 1];
        b[0] = Wcol[(size_t)k * E_OUT];
        b[1] = Wcol[(size_t)(k + 1) * E_OUT];
        acc = __builtin_amdgcn_wmma_f32_16x16x4_f32(false, a, false, b,
                                                    (short)0, acc, false, false);
    }
#pragma unroll
    for (int i = 0; i < 8; ++i) {
        const int m = m0 + i + 8 * kh;
        const int n = n0 + row;
        const int u = u_idx[m], v = v_idx[m];
        out[(size_t)m * E_OUT + n] = acc[i] + bl[n]
                                   + he_u[(size_t)u * E_OUT + n]
                                   + he_v[(size_t)v * E_OUT + n];
    }
}

// ---------------------------------------------------------------- edge streaming passes
// MODE 0: seg-max of x = h_xe * att[dst] via ordered-key global_atomic_max_u32
// MODE 1: seg-sum of exp(x - max[dst])       (global_atomic_add_f32)
// MODE 2: seg-sum of h_xe * exp(x - max[dst]) into agg (normalized later in GEMM)
// h_xe[j] = j<128 ? heN[src*128+j] : e_feat[e*64 + j-128]
// One thread = one float4 chunk; 48 chunks per edge (192 feats).
template <int MODE>
__global__ __launch_bounds__(256) void edge_pass(
        const float* __restrict__ heN, const float* __restrict__ ef,
        const float* __restrict__ att,
        const int* __restrict__ src, const int* __restrict__ dst,
        float* __restrict__ smax, float* __restrict__ ssum, float* __restrict__ agg) {
    const unsigned t = blockIdx.x * blockDim.x + threadIdx.x;  // E*48 = 24M threads
    const unsigned e = t / 48u;
    const unsigned q = t - e * 48u;
    if (e >= E_CNT) return;
    const int j = (int)q * 4;
    const int s = src[e], d = dst[e];

    const float4 h = (j < E_OUT)
        ? *(const float4*)(heN + (size_t)s * E_OUT + j)
        : *(const float4*)(ef  + (size_t)e * E_IN + (j - E_OUT));
    const size_t o = (size_t)d * ATT_D + j;
    const float4 a = *(const float4*)(att + o);

    const float x0 = h.x * a.x, x1 = h.y * a.y, x2 = h.z * a.z, x3 = h.w * a.w;

    if (MODE == 0) {
        unsigned* sm = (unsigned*)smax + o;
        atomicMax(sm + 0, f32_key(x0));
        atomicMax(sm + 1, f32_key(x1));
        atomicMax(sm + 2, f32_key(x2));
        atomicMax(sm + 3, f32_key(x3));
    } else {
        const uint4 mk = *(const uint4*)((const unsigned*)smax + o);
        const float e0 = __expf(x0 - key_f32(mk.x));
        const float e1 = __expf(x1 - key_f32(mk.y));
        const float e2 = __expf(x2 - key_f32(mk.z));
        const float e3 = __expf(x3 - key_f32(mk.w));
        if (MODE == 1) {
            atomicAdd(ssum + o + 0, e0);
            atomicAdd(ssum + o + 1, e1);
            atomicAdd(ssum + o + 2, e2);
            atomicAdd(ssum + o + 3, e3);
        } else {
            atomicAdd(agg + o + 0, h.x * e0);
            atomicAdd(agg + o + 1, h.y * e1);
            atomicAdd(agg + o + 2, h.z * e2);
            atomicAdd(agg + o + 3, h.w * e3);
        }
    }
}

// ---------------------------------------------------------------- launch
extern "C" void kernel_launch(void* const* d_in, const int* in_sizes, int n_in,
                              void* d_out, int out_size, void* d_ws, size_t ws_size,
                              hipStream_t stream) {
    const float* e_feat   = (const float*)d_in[0];
    const float* u_feat   = (const float*)d_in[1];
    const float* v_feat   = (const float*)d_in[2];
    const int*   u_idx    = (const int*)  d_in[3];
    const int*   v_idx    = (const int*)  d_in[4];
    const float* e_lin_w  = (const float*)d_in[5];
    const float* e_lin_b  = (const float*)d_in[6];
    const float* u_lin_w  = (const float*)d_in[7];
    const float* u_lin_b  = (const float*)d_in[8];
    const float* v_lin_w  = (const float*)d_in[9];
    const float* v_lin_b  = (const float*)d_in[10];
    const float* attn_u_w = (const float*)d_in[11];
    const float* attn_u_b = (const float*)d_in[12];
    const float* attn_v_w = (const float*)d_in[13];
    const float* attn_v_b = (const float*)d_in[14];
    const float* W_u_w    = (const float*)d_in[15];
    const float* W_u_b    = (const float*)d_in[16];
    const float* W_v_w    = (const float*)d_in[17];
    const float* W_v_b    = (const float*)d_in[18];
    const float* Vu_w     = (const float*)d_in[19];
    const float* Vu_b     = (const float*)d_in[20];
    const float* Vv_w     = (const float*)d_in[21];
    const float* Vv_b     = (const float*)d_in[22];

    float* out_he = (float*)d_out;                       // [E, 128]
    float* out_hu = out_he + (size_t)E_CNT * E_OUT;      // [N_U, 128]
    float* out_hv = out_hu + (size_t)NU_NODES * 128;     // [N_V, 128]

    // workspace (all fp32): 243 MB total
    float* ws    = (float*)d_ws;
    float* he_u  = ws;                                   // [N_U,128]
    float* he_v  = he_u  + (size_t)NU_NODES * E_OUT;     // [N_V,128]
    float* att_u = he_v  + (size_t)NV_NODES * E_OUT;     // [N_U,192]
    float* att_v = att_u + (size_t)NU_NODES * ATT_D;     // [N_V,192]
    float* smax  = att_v + (size_t)NV_NODES * ATT_D;     // [N,192] keys, reused both dirs
    float* ssum  = smax  + (size_t)NU_NODES * ATT_D;
    float* agg   = ssum  + (size_t)NU_NODES * ATT_D;

    const int BLK = 256;
    auto gblocks = [](int M, int N) { return ((M / 16) * (N / 16) + 7) / 8; };

    // --- node linear layers (WMMA f32) ---
    gemm_wmma_f32<128, 128, false><<<gblocks(NU_NODES, 128), BLK, 0, stream>>>(
        u_feat, nullptr, u_lin_w, u_lin_b, he_u, NU_NODES, 128, 0);
    gemm_wmma_f32<128, 128, false><<<gblocks(NV_NODES, 128), BLK, 0, stream>>>(
        v_feat, nullptr, v_lin_w, v_lin_b, he_v, NV_NODES, 128, 0);
    gemm_wmma_f32<128, 192, false><<<gblocks(NU_NODES, 192), BLK, 0, stream>>>(
        u_feat, nullptr, attn_u_w, attn_u_b, att_u, NU_NODES, 192, 0);
    gemm_wmma_f32<128, 192, false><<<gblocks(NV_NODES, 192), BLK, 0, stream>>>(
        v_feat, nullptr, attn_v_w, attn_v_b, att_v, NV_NODES, 192, 0);

    // --- he output: edge GEMM fused with endpoint gathers ---
    edge_he_wmma<<<(E_CNT / 16 * 8 + 7) / 8, BLK, 0, stream>>>(
        e_feat, e_lin_w, e_lin_b, he_u, he_v, u_idx, v_idx, out_he);

    const int segN  = NU_NODES * ATT_D;
    const int fillB = (segN + BLK - 1) / BLK;
    const int epB   = (int)(((long long)E_CNT * 48 + BLK - 1) / BLK);

    // --- backward direction (v -> u, segments by u_idx) ---
    fill_u32<<<fillB, BLK, 0, stream>>>((unsigned*)smax, NEG_INF_KEY, segN);
    fill_u32<<<fillB, BLK, 0, stream>>>((unsigned*)ssum, 0u, segN);
    fill_u32<<<fillB, BLK, 0, stream>>>((unsigned*)agg,  0u, segN);
    edge_pass<0><<<epB, BLK, 0, stream>>>(he_v, e_feat, att_u, v_idx, u_idx, smax, ssum, agg);
    edge_pass<1><<<epB, BLK, 0, stream>>>(he_v, e_feat, att_u, v_idx, u_idx, smax, ssum, agg);
    edge_pass<2><<<epB, BLK, 0, stream>>>(he_v, e_feat, att_u, v_idx, u_idx, smax, ssum, agg);
    // hu = [u_feat@Vu + b | (agg/ssum)@W_u + b]
    gemm_wmma_f32<128, 64, false><<<gblocks(NU_NODES, 64), BLK, 0, stream>>>(
        u_feat, nullptr, Vu_w, Vu_b, out_hu, NU_NODES, 128, 0);
    gemm_wmma_f32<192, 64, true><<<gblocks(NU_NODES, 64), BLK, 0, stream>>>(
        agg, ssum, W_u_w, W_u_b, out_hu, NU_NODES, 128, 64);

    // --- forward direction (u -> v, segments by v_idx), reuse buffers ---
    fill_u32<<<fillB, BLK, 0, stream>>>((unsigned*)smax, NEG_INF_KEY, segN);
    fill_u32<<<fillB, BLK, 0, stream>>>((unsigned*)ssum, 0u, segN);
    fill_u32<<<fillB, BLK, 0, stream>>>((unsigned*)agg,  0u, segN);
    edge_pass<0><<<epB, BLK, 0, stream>>>(he_u, e_feat, att_v, u_idx, v_idx, smax, ssum, agg);
    edge_pass<1><<<epB, BLK, 0, stream>>>(he_u, e_feat, att_v, u_idx, v_idx, smax, ssum, agg);
    edge_pass<2><<<epB, BLK, 0, stream>>>(he_u, e_feat, att_v, u_idx, v_idx, smax, ssum, agg);
    // hv = [v_feat@Vv + b | (agg/ssum)@W_v + b]
    gemm_wmma_f32<128, 64, false><<<gblocks(NV_NODES, 64), BLK, 0, stream>>>(
        v_feat, nullptr, Vv_w, Vv_b, out_hv, NV_NODES, 128, 0);
    gemm_wmma_f32<192, 64, true><<<gblocks(NV_NODES, 64), BLK, 0, stream>>>(
        agg, ssum, W_v_w, W_v_b, out_hv, NV_NODES, 128, 64);
}